// YoloLayer_70325794504996
// MI455X (gfx1250) — compile-verified
//
#include <hip/hip_runtime.h>
#include <cstdint>

constexpr int   kAttrib   = 85;
constexpr int   kNA       = 3;
constexpr int   kDim      = 76;               // nH == nW
constexpr int   kCH       = kNA * kAttrib;    // 255 channels
constexpr int   kWT       = 38;               // w-tile (76 = 2*38)
constexpr int   kTileElems = kCH * kWT;       // 9690 floats = 38,760 B LDS
constexpr int   kThreads  = 256;
constexpr float kStride   = 8.0f;             // 608 / 76

__global__ __launch_bounds__(kThreads)
void yolo_head_kernel(const float* __restrict__ x, float* __restrict__ out) {
    __shared__ float tile[kTileElems];

    const int wt  = blockIdx.x;   // 0..1   w-tile
    const int h   = blockIdx.y;   // 0..75
    const int b   = blockIdx.z;   // batch
    const int tid = threadIdx.x;
    const int w0  = wt * kWT;

    const int plane = kDim * kDim; // 5776
    const float* xb = x + (size_t)b * kCH * plane + h * kDim + w0;
    const unsigned ldsBase = (unsigned)(uintptr_t)&tile[0]; // LDS byte offset

    // ---- Phase 1: async DMA global -> LDS (ASYNCcnt path, gfx1250) ----
    // LDS linear index i == ch*kWT + w, i.e. the transposed-staging layout
    // falls out of the load order for free. LDS write stride across lanes is
    // 1 dword -> trivially conflict-free; global side is 128B/wave-instr.
    #pragma unroll 1
    for (int it = 0; it < (kTileElems + kThreads - 1) / kThreads; ++it) {
        int i = it * kThreads + tid;
        if (i < kTileElems) {
            int ch = i / kWT;
            int w  = i - ch * kWT;
            const float* gp = xb + (size_t)ch * plane + w;
            unsigned lds = ldsBase + (unsigned)i * 4u;
            asm volatile("global_load_async_to_lds_b32 %0, %1, off"
                         :: "v"(lds), "v"((uint64_t)(uintptr_t)gp)
                         : "memory");
        }
    }
    asm volatile("s_wait_asynccnt 0x0" ::: "memory");
    __syncthreads();

    // ---- Phase 2: transform + fully coalesced, non-temporal stores ----
    // Output tile = 9690 contiguous floats at [b][h][w0..w0+37][a][c].
    // LDS read stride across lanes = 38 floats = 2*19; 19 coprime to 32 ->
    // 32 distinct banks -> conflict-free.
    float* op = out + (size_t)b * plane * kCH + (size_t)(h * kDim + w0) * kCH;
    #pragma unroll 1
    for (int j = tid; j < kTileElems; j += kThreads) {
        int w = j / kCH;              // 0..37 (relative w)
        int r = j - w * kCH;          // a*85 + c
        int a = r / kAttrib;          // 0..2
        int c = r - a * kAttrib;      // 0..84
        float v = tile[r * kWT + w];

        bool isWH = (c == 2) | (c == 3);
        float e   = __expf(isWH ? v : -v);                 // single v_exp_f32
        float sig = __builtin_amdgcn_rcpf(1.0f + e);       // v_rcp_f32

        float anch = (c == 2) ? ((a == 0) ? 10.0f : (a == 1) ? 16.0f : 33.0f)
                              : ((a == 0) ? 13.0f : (a == 1) ? 30.0f : 23.0f);
        float res;
        if (isWH)        res = e * anch;                          // wh: exp*ANCHOR
        else if (c == 0) res = (sig + (float)(w0 + w)) * kStride; // bx
        else if (c == 1) res = (sig + (float)h) * kStride;        // by
        else             res = sig;                               // conf/cls
        __builtin_nontemporal_store(res, &op[j]);                 // th:NT stream-out
    }
}

extern "C" void kernel_launch(void* const* d_in, const int* in_sizes, int n_in,
                              void* d_out, int out_size, void* d_ws, size_t ws_size,
                              hipStream_t stream) {
    const float* x = (const float*)d_in[0];
    float* out = (float*)d_out;
    const int plane = kDim * kDim;
    int nB = in_sizes[0] / (kCH * plane);     // 32 for the reference shapes
    dim3 grid(kDim / kWT, kDim, nB);
    dim3 block(kThreads);
    yolo_head_kernel<<<grid, block, 0, stream>>>(x, out);
    (void)n_in; (void)out_size; (void)d_ws; (void)ws_size;
}